// LoraEmbedding_72335839199662
// MI455X (gfx1250) — compile-verified
//
#include <hip/hip_runtime.h>

// LoRA embedding: out[t,:] = weight[id_t,:] + SCALING * (lora_B[id_t,:] @ lora_A)
// ids: [16384] i32, weight: [50257,1024] f32, lora_A: [16,1024] f32, lora_B: [50257,16] f32
// Strategy: group 16 tokens -> 16x16 (tokens x rank) A-matrix from gathered lora_B rows,
// multiply against 16-column tiles of lora_A with V_WMMA_F32_16X16X4_F32 (4x, K=16),
// fuse the weight-row gather + scale + store. Memory-bound: ~132 MB total traffic.

typedef __attribute__((ext_vector_type(2))) float v2f;
typedef __attribute__((ext_vector_type(8))) float v8f;

#define DIM 1024
#define RANK 16
#define SCALING 2.0f   // ALPHA / RANK = 32/16

__global__ __launch_bounds__(256) void lora_embed_wmma(
    const int*   __restrict__ ids,
    const float* __restrict__ weight,
    const float* __restrict__ lora_A,
    const float* __restrict__ lora_B,
    float*       __restrict__ out)
{
    const int lane = threadIdx.x & 31;
    const int wave = threadIdx.x >> 5;
    const int g    = blockIdx.x * 8 + wave;   // 16-token group index
    const int hi   = lane >> 4;               // 0: lanes 0-15, 1: lanes 16-31
    const int lo   = lane & 15;

    // Token id this lane owns (m = lane & 15; both wave halves mirror the same ids).
    const int id_m = ids[g * 16 + lo];

    // ---- A fragment: 16x16 (M=token x K=rank) of lora_B rows, split into 4 K-chunks of 4.
    // Layout (32-bit A 16x4): lane L: M = L%16, VGPR0 = K = 4c + 2*(L/16), VGPR1 = +1.
    v2f a[4];
#pragma unroll
    for (int c = 0; c < 4; ++c) {
        const int kb = 4 * c + 2 * hi;
        a[c].x = lora_B[id_m * RANK + kb];
        a[c].y = lora_B[id_m * RANK + kb + 1];
    }

    // ---- Row ids / output rows matching the C/D layout:
    // D VGPR r holds M=r for lanes 0-15 and M=r+8 for lanes 16-31.
    int row_id[8];
    int row_tok[8];
#pragma unroll
    for (int r = 0; r < 8; ++r) {
        const int m = r + 8 * hi;
        row_id[r]  = __shfl(id_m, m, 32);     // ds_bpermute broadcast of token ids
        row_tok[r] = g * 16 + m;
    }

    // ---- Loop over 16-column tiles in this wave's column quarter.
    const int tile0 = blockIdx.y * 16;
    for (int tt = 0; tt < 16; ++tt) {
        const int col0 = (tile0 + tt) * 16;

        // B fragment: 4x16 (K x N) tile of lora_A per chunk.
        // VGPR0: lanes 0-15 = row K=4c (N=lane), lanes 16-31 = row K=4c+2; VGPR1 = +1.
        v2f b[4];
#pragma unroll
        for (int c = 0; c < 4; ++c) {
            const int k = 4 * c + 2 * hi;
            b[c].x = lora_A[k * DIM + col0 + lo];
            b[c].y = lora_A[(k + 1) * DIM + col0 + lo];
        }

        // D = A x B (+C), accumulated over K=16 via four K=4 WMMAs, full f32.
        v8f acc = {};
#pragma unroll
        for (int c = 0; c < 4; ++c) {
            acc = __builtin_amdgcn_wmma_f32_16x16x4_f32(
                /*neg_a=*/false, a[c], /*neg_b=*/false, b[c],
                /*c_mod=*/(short)0, acc, /*reuse_a=*/false, /*reuse_b=*/false);
        }

        // Fused epilogue: gather weight rows (C-layout addressing), scale, store.
#pragma unroll
        for (int r = 0; r < 8; ++r) {
            const float w = weight[row_id[r] * DIM + col0 + lo];
            out[row_tok[r] * DIM + col0 + lo] = w + SCALING * acc[r];
        }
    }
}

extern "C" void kernel_launch(void* const* d_in, const int* in_sizes, int n_in,
                              void* d_out, int out_size, void* d_ws, size_t ws_size,
                              hipStream_t stream) {
    const int*   ids    = (const int*)d_in[0];
    const float* weight = (const float*)d_in[1];
    const float* lora_A = (const float*)d_in[2];
    const float* lora_B = (const float*)d_in[3];
    float*       out    = (float*)d_out;

    const int n_tokens = in_sizes[0];        // 16384
    const int groups   = n_tokens / 16;      // 1024 (divides evenly for this shape)

    dim3 block(256);                         // 8 waves / block (wave32)
    dim3 grid(groups / 8, 4);                // x: group octets, y: 256-column quarters
    lora_embed_wmma<<<grid, block, 0, stream>>>(ids, weight, lora_A, lora_B, out);
}